// MCGRULayer_69037304316198
// MI455X (gfx1250) — compile-verified
//
#include <hip/hip_runtime.h>

typedef float v2f __attribute__((ext_vector_type(2)));
typedef float v8f __attribute__((ext_vector_type(8)));

#define B_    256
#define T_    512
#define D_    64
#define F_    8
#define G_    24          // 3*F
#define HID_  32
#define KOUT_ (D_*F_)     // 512
#define WOSTR (HID_+KOUT_) // 544 = W_out row stride

// ---------------------------------------------------------------------------
// Kernel 1: fold demo through W_out:
//   demo[b,j] = b_static[j] + sum_k static[b,k]*W_static[j,k]
//   base[b,n] = b_out[n]   + sum_j demo[b,j]*W_out[n,j]
// Tiny (256x32); one wave per batch row.
// ---------------------------------------------------------------------------
__global__ void __launch_bounds__(HID_) k1_base(
    const float* __restrict__ stat, const float* __restrict__ Ws,
    const float* __restrict__ bs,   const float* __restrict__ Wo,
    const float* __restrict__ bo,   float* __restrict__ base) {
  __shared__ float demo[HID_];
  const int b = blockIdx.x, j = threadIdx.x;
  float acc = bs[j];
#pragma unroll
  for (int k = 0; k < 32; ++k) acc += stat[b*32 + k] * Ws[j*32 + k];
  demo[j] = acc;
  __syncthreads();
  float o = bo[j];
#pragma unroll
  for (int k = 0; k < HID_; ++k) o += demo[k] * Wo[j*WOSTR + k];
  base[b*HID_ + j] = o;
}

// ---------------------------------------------------------------------------
// Kernel 2: xl = x @ W_lab^T + b_lab  -- [131072,64]x[64,64] fp32 GEMM.
// One wave owns a 16-row tile and all 4 N-tiles (N=64), K=64 in 16 k4-steps.
// V_WMMA_F32_16X16X4_F32 layouts:
//   A 16x4: lane m=lane&15; VGPR0/1 = K {0,1} (lanes 0-15) or {2,3} (16-31)
//   B 4x16: lane n=lane&15; same K split  -> B[k][n] = W_lab[n][k]
//   C 16x16: VGPR j = row j (lanes 0-15) / j+8 (lanes 16-31), col = lane&15
// ---------------------------------------------------------------------------
__global__ void __launch_bounds__(128) k2_xlab(
    const float* __restrict__ x, const float* __restrict__ Wl,
    const float* __restrict__ bl, float* __restrict__ xl) {
  const int lane = threadIdx.x & 31;
  const int tile = blockIdx.x * 4 + (threadIdx.x >> 5);
  const int rowBase = tile * 16;
  const int m    = lane & 15;
  const int half = lane >> 4;
  const float* arow = x + (size_t)(rowBase + m) * D_;
  v8f c[4] = {{}, {}, {}, {}};
#pragma unroll
  for (int k0 = 0; k0 < D_; k0 += 4) {
    v2f a = *(const v2f*)(arow + k0 + 2*half);
#pragma unroll
    for (int nb = 0; nb < 4; ++nb) {
      v2f bm = *(const v2f*)(Wl + (size_t)(nb*16 + m) * D_ + k0 + 2*half);
      c[nb] = __builtin_amdgcn_wmma_f32_16x16x4_f32(
          false, a, false, bm, (short)0, c[nb], false, false);
    }
  }
#pragma unroll
  for (int nb = 0; nb < 4; ++nb) {
    const float bias = bl[nb*16 + m];
#pragma unroll
    for (int j = 0; j < 8; ++j) {
      const int row = rowBase + j + 8*half;
      xl[(size_t)row * D_ + nb*16 + m] = c[nb][j] + bias;
    }
  }
}

// ---------------------------------------------------------------------------
// Kernel 3: per-(b,d) GRU scan over t.  Block = 4 batches x 32 channels
// (each wave: one batch, lanes = 32 consecutive channels -> coalesced xl
// reads (128B/wave/step) and coalesced chan writes (1KB/wave/step)).
// W_hh cached in LDS with padded stride 196 floats (196 % 64 == 4) so the
// per-lane float4 fetches rotate across bank groups.
// ---------------------------------------------------------------------------
#define NCH 32
#define NB  4
#define WSTRIDE 196

__global__ void __launch_bounds__(NCH*NB) k3_gru(
    const float* __restrict__ xl,  const float* __restrict__ Wih,
    const float* __restrict__ Whh, const float* __restrict__ bih,
    const float* __restrict__ bhh, float* __restrict__ chan) {
  __shared__ __align__(16) float sW[NCH * WSTRIDE];
  __shared__ float sWih[G_ * NCH];
  __shared__ float sBih[G_ * NCH];
  __shared__ float sBhh[G_ * NCH];

  const int tid = threadIdx.x;
  const int d0 = blockIdx.y * NCH;
  const int b0 = blockIdx.x * NB;

  for (int i = tid; i < NCH * (G_*F_); i += NCH*NB) {
    const int dl = i / (G_*F_), r = i % (G_*F_);
    sW[dl*WSTRIDE + r] = Whh[(size_t)(d0 + dl) * (G_*F_) + r];
  }
  for (int i = tid; i < G_*NCH; i += NCH*NB) {
    const int g = i / NCH, dl = i % NCH;
    sWih[g*NCH + dl] = Wih[(d0 + dl)*G_ + g];
    sBih[g*NCH + dl] = bih[(d0 + dl)*G_ + g];
    sBhh[g*NCH + dl] = bhh[(d0 + dl)*G_ + g];
  }
  __syncthreads();

  const int dl = tid & (NCH - 1);
  const int b  = b0 + (tid / NCH);
  const int d  = d0 + dl;
  const float* wrow = sW + dl*WSTRIDE;

  float h[F_];
#pragma unroll
  for (int f = 0; f < F_; ++f) h[f] = 0.f;

  for (int t = 0; t < T_; ++t) {
    const float xv = xl[(size_t)(b*T_ + t) * D_ + d];
    float acc[G_];
#pragma unroll
    for (int g = 0; g < G_; ++g) {           // gh[g] = h . W_hh[d][g][:]
      const float4 w0 = *(const float4*)(wrow + g*F_);
      const float4 w1 = *(const float4*)(wrow + g*F_ + 4);
      acc[g] = h[0]*w0.x + h[1]*w0.y + h[2]*w0.z + h[3]*w0.w
             + h[4]*w1.x + h[5]*w1.y + h[6]*w1.z + h[7]*w1.w;
    }
    float hn_[F_];
#pragma unroll
    for (int f = 0; f < F_; ++f) {
      const float ir = fmaf(xv, sWih[f*NCH + dl],        sBih[f*NCH + dl]);
      const float iz = fmaf(xv, sWih[(F_+f)*NCH + dl],   sBih[(F_+f)*NCH + dl]);
      const float in_= fmaf(xv, sWih[(2*F_+f)*NCH + dl], sBih[(2*F_+f)*NCH + dl]);
      const float hr = acc[f]        + sBhh[f*NCH + dl];
      const float hz = acc[F_+f]     + sBhh[(F_+f)*NCH + dl];
      const float hn = acc[2*F_+f]   + sBhh[(2*F_+f)*NCH + dl];
      const float r  = 1.f / (1.f + __expf(-(ir + hr)));
      const float z  = 1.f / (1.f + __expf(-(iz + hz)));
      const float nx = in_ + r*hn;
      const float e  = __expf(2.f*nx);
      const float n  = 1.f - 2.f/(e + 1.f);       // tanh(nx), saturates correctly
      hn_[f] = (1.f - z)*n + z*h[f];
    }
#pragma unroll
    for (int f = 0; f < F_; ++f) h[f] = hn_[f];
    float* dst = chan + (size_t)(b*T_ + t) * KOUT_ + d*F_;
    *(float4*)(dst)     = make_float4(h[0], h[1], h[2], h[3]);
    *(float4*)(dst + 4) = make_float4(h[4], h[5], h[6], h[7]);
  }
}

// ---------------------------------------------------------------------------
// Kernel 4: out[row,n] = base[b(row),n] + sum_k chan[row,k]*W_out[n, 32+k]
// [131072,512]x[512,32] fp32 GEMM; wave owns a 16-row tile, both N-tiles.
// ---------------------------------------------------------------------------
__global__ void __launch_bounds__(128) k4_out(
    const float* __restrict__ chan, const float* __restrict__ Wo,
    const float* __restrict__ base, float* __restrict__ out) {
  const int lane = threadIdx.x & 31;
  const int tile = blockIdx.x * 4 + (threadIdx.x >> 5);
  const int rowBase = tile * 16;
  const int m    = lane & 15;
  const int half = lane >> 4;
  const float* arow = chan + (size_t)(rowBase + m) * KOUT_;
  const float* b0p  = Wo + (size_t)m        * WOSTR + HID_;   // n = m
  const float* b1p  = Wo + (size_t)(16 + m) * WOSTR + HID_;   // n = 16+m
  v8f c0 = {}, c1 = {};
#pragma unroll 4
  for (int k0 = 0; k0 < KOUT_; k0 += 4) {
    v2f a   = *(const v2f*)(arow + k0 + 2*half);
    v2f bb0 = *(const v2f*)(b0p  + k0 + 2*half);
    v2f bb1 = *(const v2f*)(b1p  + k0 + 2*half);
    c0 = __builtin_amdgcn_wmma_f32_16x16x4_f32(false, a, false, bb0, (short)0, c0, false, false);
    c1 = __builtin_amdgcn_wmma_f32_16x16x4_f32(false, a, false, bb1, (short)0, c1, false, false);
  }
#pragma unroll
  for (int j = 0; j < 8; ++j) {
    const int row = rowBase + j + 8*half;
    const int bb  = row >> 9;                 // row / T_
    out[(size_t)row*HID_ + m]      = c0[j] + base[bb*HID_ + m];
    out[(size_t)row*HID_ + 16 + m] = c1[j] + base[bb*HID_ + 16 + m];
  }
}

// ---------------------------------------------------------------------------
extern "C" void kernel_launch(void* const* d_in, const int* in_sizes, int n_in,
                              void* d_out, int out_size, void* d_ws, size_t ws_size,
                              hipStream_t stream) {
  const float* x    = (const float*)d_in[0];
  const float* stat = (const float*)d_in[1];
  const float* Ws   = (const float*)d_in[2];
  const float* bs   = (const float*)d_in[3];
  const float* Wl   = (const float*)d_in[4];
  const float* bl   = (const float*)d_in[5];
  const float* Wih  = (const float*)d_in[6];
  const float* Whh  = (const float*)d_in[7];
  const float* bih  = (const float*)d_in[8];
  const float* bhh  = (const float*)d_in[9];
  const float* Wo   = (const float*)d_in[10];
  const float* bo   = (const float*)d_in[11];
  float* out = (float*)d_out;

  // scratch layout: base (32KB) | xl (33.5MB) | chan (256MB)
  char*  ws   = (char*)d_ws;
  float* base = (float*)ws;
  float* xl   = (float*)(ws + 32*1024);
  float* chan = (float*)(ws + 32*1024 + sizeof(float)*(size_t)B_*T_*D_);

  k1_base<<<B_, HID_, 0, stream>>>(stat, Ws, bs, Wo, bo, base);
  k2_xlab<<<(B_*T_/16)/4, 128, 0, stream>>>(x, Wl, bl, xl);
  dim3 g3(B_/NB, D_/NCH);
  k3_gru<<<g3, NCH*NB, 0, stream>>>(xl, Wih, Whh, bih, bhh, chan);
  k4_out<<<(B_*T_/16)/4, 128, 0, stream>>>(chan, Wo, base, out);
}